// st_network_66898410602732
// MI455X (gfx1250) — compile-verified
//
#include <hip/hip_runtime.h>
#include <hip/hip_bf16.h>

typedef __attribute__((ext_vector_type(2))) float v2f;
typedef __attribute__((ext_vector_type(8))) float v8f;

#define FEAT 128

// ---------------------------------------------------------------------------
// Generic WMMA f32 GEMM:  OUT[M,N] = act( (X (+X2)) @ W (+bias) )
// X: [M,K] row-major, W: [K,N] row-major, OUT: [M,N] row-major.
// One wave (32 threads) per 16x16 output tile. grid = (ceil(M/16), N/16).
// V_WMMA_F32_16X16X4_F32, K stepped by 4, unrolled 4x (16 MACs per group).
// HAS_X2: fuse elementwise X+X2 into the A-fragment load (pre-MLP).
// FUSED : apply bias + ReLU on store.
// ---------------------------------------------------------------------------
template <bool HAS_X2, bool FUSED>
__global__ void __launch_bounds__(32)
wmma_gemm_f32(const float* __restrict__ X, const float* __restrict__ X2,
              const float* __restrict__ W, const float* __restrict__ bias,
              float* __restrict__ OUT, int M, int K, int N)
{
    const int lane = threadIdx.x;          // 0..31, EXEC all ones throughout
    const int half = lane >> 4;            // 0 or 1
    const int l16  = lane & 15;
    const int m0   = blockIdx.x * 16;
    const int n0   = blockIdx.y * 16;

    // A-fragment row for this lane (clamped; garbage only feeds output rows
    // >= M which are never stored — row m of A affects only row m of D).
    int arow = m0 + l16;
    if (arow >= M) arow = M - 1;

    // Hoisted base pointers; loop advances by constant strides so the
    // backend folds addressing into load offsets.
    const float* __restrict__ ap  = X + (size_t)arow * K + half * 2;   // 8B aligned
    const float* __restrict__ ap2 = HAS_X2 ? X2 + (size_t)arow * K + half * 2
                                           : nullptr;
    const int col = n0 + l16;              // B / D column for this lane
    const float* __restrict__ bp  = W + (size_t)(half * 2) * N + col;
    const size_t bstep = (size_t)4 * N;

    v8f acc = {};
#pragma unroll 4
    for (int kb = 0; kb < K; kb += 4) {
        v2f a = *(const v2f*)ap;
        if (HAS_X2) {
            v2f a2 = *(const v2f*)ap2;
            a.x += a2.x; a.y += a2.y;
            ap2 += 4;
        }
        v2f b;
        b.x = bp[0];
        b.y = bp[N];
        ap += 4;
        bp += bstep;
        acc = __builtin_amdgcn_wmma_f32_16x16x4_f32(
            false, a, false, b, (short)0, acc, false, false);
    }

    // C/D layout: lanes 0-15 -> rows m0+0..7, lanes 16-31 -> rows m0+8..15,
    // col = n0 + lane%16.
    const float bv = FUSED ? bias[col] : 0.0f;
#pragma unroll
    for (int i = 0; i < 8; ++i) {
        const int m = m0 + half * 8 + i;
        if (m < M) {
            float v = acc[i];
            if (FUSED) v = fmaxf(v + bv, 0.0f);
            OUT[(size_t)m * N + col] = v;
        }
    }
}

// ---------------------------------------------------------------------------
// Degree / normalization
// ---------------------------------------------------------------------------
__global__ void deg_init(float* __restrict__ deg, int n) {
    int i = blockIdx.x * blockDim.x + threadIdx.x;
    if (i < n) deg[i] = 1.0f;              // self-loop contribution
}
__global__ void deg_accum(const int* __restrict__ dst, float* __restrict__ deg, int E) {
    int e = blockIdx.x * blockDim.x + threadIdx.x;
    if (e < E) atomicAdd(&deg[dst[e]], 1.0f);
}
__global__ void deg_rsqrt(float* __restrict__ deg, int n) {
    int i = blockIdx.x * blockDim.x + threadIdx.x;
    if (i < n) deg[i] = rsqrtf(deg[i]);    // deg >= 1 always (self-loops)
}

__global__ void zero_f32(float* __restrict__ p, int n) {
    int i = blockIdx.x * blockDim.x + threadIdx.x;
    if (i < n) p[i] = 0.0f;
}

// ---------------------------------------------------------------------------
// Edge scatter: one wave per edge, float4 per lane (128 features).
// agg[d] += t[s] * dinv[s]*dinv[d]
// ---------------------------------------------------------------------------
__global__ void __launch_bounds__(256)
scatter_edges(const int* __restrict__ src, const int* __restrict__ dst,
              const float* __restrict__ dinv, const float* __restrict__ t,
              float* __restrict__ agg, int E)
{
    int e    = blockIdx.x * (blockDim.x >> 5) + (threadIdx.x >> 5);
    int lane = threadIdx.x & 31;
    if (e >= E) return;
    int s = src[e], d = dst[e];
    float w = dinv[s] * dinv[d];
    const float4* tp = (const float4*)(t + (size_t)s * FEAT);
    float4 v = tp[lane];
    float* ap = agg + (size_t)d * FEAT + lane * 4;
    atomicAdd(ap + 0, v.x * w);
    atomicAdd(ap + 1, v.y * w);
    atomicAdd(ap + 2, v.z * w);
    atomicAdd(ap + 3, v.w * w);
}

// Self-loop (t[n]*dinv[n]^2) + bias + ReLU, in place on agg.
__global__ void conv_finalize(float* __restrict__ agg, const float* __restrict__ t,
                              const float* __restrict__ dinv,
                              const float* __restrict__ bias, int n)
{
    int i = blockIdx.x * blockDim.x + threadIdx.x;
    if (i >= n * FEAT) return;
    int node = i >> 7, f = i & (FEAT - 1);
    float dv = dinv[node];
    float v = agg[i] + t[i] * dv * dv + bias[f];
    agg[i] = fmaxf(v, 0.0f);
}

// ---------------------------------------------------------------------------
// Mean pool: one wave per node, float4 per lane; counts from lane 0.
// ---------------------------------------------------------------------------
__global__ void __launch_bounds__(256)
pool_accum(const float* __restrict__ x, const int* __restrict__ batch,
           float* __restrict__ sums, float* __restrict__ cnts, int n)
{
    int node = blockIdx.x * (blockDim.x >> 5) + (threadIdx.x >> 5);
    int lane = threadIdx.x & 31;
    if (node >= n) return;
    int g = batch[node];
    const float4* xp = (const float4*)(x + (size_t)node * FEAT);
    float4 v = xp[lane];
    float* sp = sums + (size_t)g * FEAT + lane * 4;
    atomicAdd(sp + 0, v.x);
    atomicAdd(sp + 1, v.y);
    atomicAdd(sp + 2, v.z);
    atomicAdd(sp + 3, v.w);
    if (lane == 0) atomicAdd(&cnts[g], 1.0f);
}

__global__ void pool_div(float* __restrict__ sums, const float* __restrict__ cnts, int G) {
    int i = blockIdx.x * blockDim.x + threadIdx.x;
    if (i < G * FEAT) sums[i] /= fmaxf(cnts[i >> 7], 1.0f);
}

// ---------------------------------------------------------------------------
// Host-side orchestration
// ---------------------------------------------------------------------------
static inline size_t align256(size_t x) { return (x + 255) & ~(size_t)255; }

extern "C" void kernel_launch(void* const* d_in, const int* in_sizes, int n_in,
                              void* d_out, int out_size, void* d_ws, size_t ws_size,
                              hipStream_t stream)
{
    const float* emb   = (const float*)d_in[0];
    const float* pos   = (const float*)d_in[1];
    const int*   eidx  = (const int*)d_in[2];
    const int*   batch = (const int*)d_in[3];
    const float* W_pre = (const float*)d_in[4];
    const float* b_pre = (const float*)d_in[5];
    const float* W_g1  = (const float*)d_in[6];
    const float* b_g1  = (const float*)d_in[7];
    const float* W_g2  = (const float*)d_in[8];
    const float* b_g2  = (const float*)d_in[9];
    const float* W_h1  = (const float*)d_in[10];
    const float* b_h1  = (const float*)d_in[11];
    const float* W_h2  = (const float*)d_in[12];
    const float* b_h2  = (const float*)d_in[13];

    const int n = in_sizes[0] / 64;        // 50000 nodes
    const int E = in_sizes[2] / 2;         // 800000 edges
    const int G = 128;                     // graphs

    const int* src = eidx;
    const int* dst = eidx + E;

    // Workspace carve-up
    char* ws = (char*)d_ws;
    size_t off = 0;
    float* bufA  = (float*)(ws + off); off = align256(off + (size_t)n * FEAT * 4);
    float* bufB  = (float*)(ws + off); off = align256(off + (size_t)n * FEAT * 4);
    float* dinv  = (float*)(ws + off); off = align256(off + (size_t)n * 4);
    float* sums  = (float*)(ws + off); off = align256(off + (size_t)G * FEAT * 4);
    float* cnts  = (float*)(ws + off); off = align256(off + (size_t)G * 4);
    float* hbuf  = (float*)(ws + off); off = align256(off + (size_t)G * 64 * 4);
    (void)ws_size;

    const int TB = 256;
    const int nf = n * FEAT;

    // 1) degrees -> dinv = rsqrt(deg)   (deg includes self-loop)
    deg_init  <<<(n + TB - 1) / TB, TB, 0, stream>>>(dinv, n);
    deg_accum <<<(E + TB - 1) / TB, TB, 0, stream>>>(dst, dinv, E);
    deg_rsqrt <<<(n + TB - 1) / TB, TB, 0, stream>>>(dinv, n);

    // 2) h0 = relu((emb + pos) @ W_pre + b_pre)     [n,128], K=64
    wmma_gemm_f32<true, true><<<dim3((n + 15) / 16, FEAT / 16), 32, 0, stream>>>(
        emb, pos, W_pre, b_pre, bufA, n, 64, FEAT);

    // 3) conv1: t = h0 @ W_g1 ; agg = scatter ; h1 = relu(agg + self + b)
    wmma_gemm_f32<false, false><<<dim3((n + 15) / 16, FEAT / 16), 32, 0, stream>>>(
        bufA, nullptr, W_g1, nullptr, bufB, n, FEAT, FEAT);
    zero_f32<<<(nf + TB - 1) / TB, TB, 0, stream>>>(bufA, nf);
    scatter_edges<<<(E + 7) / 8, TB, 0, stream>>>(src, dst, dinv, bufB, bufA, E);
    conv_finalize<<<(nf + TB - 1) / TB, TB, 0, stream>>>(bufA, bufB, dinv, b_g1, n);

    // 4) conv2
    wmma_gemm_f32<false, false><<<dim3((n + 15) / 16, FEAT / 16), 32, 0, stream>>>(
        bufA, nullptr, W_g2, nullptr, bufB, n, FEAT, FEAT);
    zero_f32<<<(nf + TB - 1) / TB, TB, 0, stream>>>(bufA, nf);
    scatter_edges<<<(E + 7) / 8, TB, 0, stream>>>(src, dst, dinv, bufB, bufA, E);
    conv_finalize<<<(nf + TB - 1) / TB, TB, 0, stream>>>(bufA, bufB, dinv, b_g2, n);

    // 5) global mean pool  (sums+cnts contiguous: zero both in one pass)
    zero_f32<<<(G * FEAT + G + TB - 1) / TB, TB, 0, stream>>>(sums, G * FEAT + G);
    pool_accum<<<(n + 7) / 8, TB, 0, stream>>>(bufA, batch, sums, cnts, n);
    pool_div<<<(G * FEAT + TB - 1) / TB, TB, 0, stream>>>(sums, cnts, G);

    // 6) head: [128,128] -> relu -> [128,64] -> relu -> [128,32] = d_out
    wmma_gemm_f32<false, true><<<dim3(G / 16, 64 / 16), 32, 0, stream>>>(
        sums, nullptr, W_h1, b_h1, hbuf, G, FEAT, 64);
    wmma_gemm_f32<false, true><<<dim3(G / 16, 32 / 16), 32, 0, stream>>>(
        hbuf, nullptr, W_h2, b_h2, (float*)d_out, G, 64, 32);

    (void)n_in; (void)out_size;
}